// SPrompt_mul_86723979641560
// MI455X (gfx1250) — compile-verified
//
#include <hip/hip_runtime.h>
#include <math.h>

// ---------------- problem constants (from reference) ----------------
#define B_      64
#define S_      2048
#define D_      768
#define L_      2
#define POOL_   30
#define POOLP_  32      // padded pool for WMMA tiles
#define LEN_    5
#define TOPK_   5
#define SSPLIT_ 16      // S-dimension split for the streaming reduction

typedef __attribute__((ext_vector_type(2))) float v2f;
typedef __attribute__((ext_vector_type(8))) float v8f;

// ---------------- Kernel A: partial mean over S (streaming, coalesced B128 loads) ----
// grid = (B_, SSPLIT_), block = 192 (each thread owns 4 contiguous d's -> full D row)
__global__ __launch_bounds__(192) void reduce_partial_kernel(
    const float* __restrict__ x, float* __restrict__ part) {
  const int b = blockIdx.x;
  const int chunk = blockIdx.y;
  const int d4 = threadIdx.x;                 // 0..191 -> float4 lane
  const int s_per = S_ / SSPLIT_;             // 128
  const float4* xb = (const float4*)(x + (size_t)b * S_ * D_);
  float4 acc = make_float4(0.f, 0.f, 0.f, 0.f);
  size_t base = (size_t)chunk * s_per * (D_ / 4) + d4;
  for (int s = 0; s < s_per; ++s) {
    float4 v = xb[base + (size_t)s * (D_ / 4)];
    acc.x += v.x; acc.y += v.y; acc.z += v.z; acc.w += v.w;
  }
  float4* p = (float4*)(part + ((size_t)chunk * B_ + b) * D_);
  p[d4] = acc;
}

// ---------------- Kernel B: combine partials -> mean -> l2 normalize x ----
// grid = B_, block = 256 (3 d's per thread)
__global__ __launch_bounds__(256) void finalize_xnorm_kernel(
    const float* __restrict__ part, float* __restrict__ xn) {
  __shared__ float red[256];
  const int b = blockIdx.x;
  const int t = threadIdx.x;
  float m[3];
  float sq = 0.f;
  for (int j = 0; j < 3; ++j) {
    const int d = t + j * 256;
    float s = 0.f;
    for (int c = 0; c < SSPLIT_; ++c)
      s += part[((size_t)c * B_ + b) * D_ + d];
    m[j] = s * (1.0f / (float)S_);
    sq += m[j] * m[j];
  }
  red[t] = sq;
  __syncthreads();
  for (int off = 128; off > 0; off >>= 1) {
    if (t < off) red[t] += red[t + off];
    __syncthreads();
  }
  const float inv = rsqrtf(fmaxf(red[0], 1e-12f));
  for (int j = 0; j < 3; ++j)
    xn[(size_t)b * D_ + t + j * 256] = m[j] * inv;
}

// ---------------- Kernel C: l2-normalize prompt keys, pad pool rows 30..31 with 0 ----
// grid = (2 types, POOLP_ rows), block = 256 (3 d's per thread)
__global__ __launch_bounds__(256) void key_norm_kernel(
    const float* __restrict__ skey, const float* __restrict__ tkey,
    float* __restrict__ skn, float* __restrict__ tkn) {
  __shared__ float red[256];
  const float* in  = (blockIdx.x == 0) ? skey : tkey;
  float* outp      = (blockIdx.x == 0) ? skn  : tkn;
  const int p = blockIdx.y;
  const int t = threadIdx.x;
  if (p >= POOL_) {   // uniform over block: zero padding rows
    for (int j = 0; j < 3; ++j) outp[(size_t)p * D_ + t + j * 256] = 0.f;
    return;
  }
  float v[3];
  float sq = 0.f;
  for (int j = 0; j < 3; ++j) {
    v[j] = in[(size_t)p * D_ + t + j * 256];
    sq += v[j] * v[j];
  }
  red[t] = sq;
  __syncthreads();
  for (int off = 128; off > 0; off >>= 1) {
    if (t < off) red[t] += red[t + off];
    __syncthreads();
  }
  const float inv = rsqrtf(fmaxf(red[0], 1e-12f));
  for (int j = 0; j < 3; ++j)
    outp[(size_t)p * D_ + t + j * 256] = v[j] * inv;
}

// ---------------- Kernel D: similarity = x_norm @ key_norm^T via V_WMMA_F32_16X16X4_F32 ----
// grid = (4 M-tiles, 2 N-tiles, 2 types), block = 32 (one wave per 16x16 tile)
// A 16x4 f32 layout (ISA 7.12.2): lanes 0-15 M=lane {K0,K1}, lanes 16-31 M=lane-16 {K2,K3}
// B 4x16 f32 mirrored across lanes; C/D: VGPR i -> M=i (lanes 0-15) / M=i+8 (lanes 16-31)
__global__ __launch_bounds__(32) void sim_wmma_kernel(
    const float* __restrict__ xn, const float* __restrict__ skn,
    const float* __restrict__ tkn, float* __restrict__ out_s,
    float* __restrict__ out_t) {
  const int m0 = blockIdx.x * 16;
  const int n0 = blockIdx.y * 16;
  const float* key = (blockIdx.z == 0) ? skn : tkn;
  float* out       = (blockIdx.z == 0) ? out_s : out_t;
  const int lane = threadIdx.x;
  const int half = lane >> 4;     // 0: K pair {0,1}, 1: K pair {2,3}
  const int lid  = lane & 15;
  const float* arow = xn  + (size_t)(m0 + lid) * D_;
  const float* brow = key + (size_t)(n0 + lid) * D_;
  v8f acc = {};
  for (int k = 0; k < D_; k += 4) {
    v2f a, b;
    a.x = arow[k + 2 * half];
    a.y = arow[k + 2 * half + 1];
    b.x = brow[k + 2 * half];
    b.y = brow[k + 2 * half + 1];
    acc = __builtin_amdgcn_wmma_f32_16x16x4_f32(
        /*neg_a=*/false, a, /*neg_b=*/false, b,
        /*c_mod=*/(short)0, acc, /*reuse_a=*/false, /*reuse_b=*/false);
  }
  const int n = n0 + lid;
  if (n < POOL_) {
    for (int i = 0; i < 8; ++i) {
      const int m = m0 + i + 8 * half;
      out[(size_t)m * POOL_ + n] = acc[i];
    }
  }
}

// ---------------- Kernel E: top-5 of 30 per (b,type); reduce_sim = sum(top5)/B ----
// grid = 2 (type), block = 64 (one thread per b). Deterministic block reduction.
__global__ __launch_bounds__(64) void topk_kernel(
    const float* __restrict__ sim_s, const float* __restrict__ sim_t,
    int* __restrict__ idx_s, int* __restrict__ idx_t,
    float* __restrict__ out_rs, float* __restrict__ out_rt) {
  __shared__ float red[64];
  const int type = blockIdx.x;
  const int b = threadIdx.x;
  const float* sim = (type == 0) ? sim_s : sim_t;
  int* idx         = (type == 0) ? idx_s : idx_t;
  float v[POOL_];
  for (int p = 0; p < POOL_; ++p) v[p] = sim[(size_t)b * POOL_ + p];
  float sum = 0.f;
  for (int k = 0; k < TOPK_; ++k) {
    float best = -INFINITY;
    int bi = 0;
    for (int p = 0; p < POOL_; ++p) {
      if (v[p] > best) { best = v[p]; bi = p; }   // strict '>' => lowest index on ties
    }
    idx[b * TOPK_ + k] = bi;
    sum += best;
    v[bi] = -INFINITY;
  }
  red[b] = sum;
  __syncthreads();
  for (int off = 32; off > 0; off >>= 1) {
    if (b < off) red[b] += red[b + off];
    __syncthreads();
  }
  if (b == 0) {
    const float r = red[0] * (1.0f / (float)B_);
    if (type == 0) *out_rs = r; else *out_rt = r;
  }
}

// ---------------- Kernel F: gather selected prompts -> [L, B, K*LEN, D] ----
// grid = (B_*TOPK_, L_, 2 types), block = 256; each block copies LEN*D = 3840 floats
__global__ __launch_bounds__(256) void gather_kernel(
    const float* __restrict__ sp, const float* __restrict__ tp,
    const int* __restrict__ idx_s, const int* __restrict__ idx_t,
    float* __restrict__ out_s, float* __restrict__ out_t) {
  const int type = blockIdx.z;
  const float* prm = (type == 0) ? sp : tp;
  const int* idx   = (type == 0) ? idx_s : idx_t;
  float* outp      = (type == 0) ? out_s : out_t;
  const int l  = blockIdx.y;
  const int bk = blockIdx.x;
  const int b  = bk / TOPK_;
  const int k  = bk % TOPK_;
  const int pi = idx[b * TOPK_ + k];
  const float4* src = (const float4*)(prm + ((size_t)l * POOL_ + pi) * LEN_ * D_);
  float4* dst = (float4*)(outp + ((((size_t)l * B_ + b) * TOPK_ + k) * LEN_) * D_);
  const int n4 = LEN_ * D_ / 4;   // 960
  for (int i = threadIdx.x; i < n4; i += 256) dst[i] = src[i];
}

// ---------------- host launcher ----------------
extern "C" void kernel_launch(void* const* d_in, const int* in_sizes, int n_in,
                              void* d_out, int out_size, void* d_ws, size_t ws_size,
                              hipStream_t stream) {
  const float* x_embed  = (const float*)d_in[0];   // [64, 2048, 768]
  const float* s_prompt = (const float*)d_in[1];   // [2, 30, 5, 768]
  const float* t_prompt = (const float*)d_in[2];   // [2, 30, 5, 768]
  const float* s_pkey   = (const float*)d_in[3];   // [30, 768]
  const float* t_pkey   = (const float*)d_in[4];   // [30, 768]

  float* out = (float*)d_out;
  // output layout: s_batched | t_batched | s_sim | t_sim | s_reduce | t_reduce
  const size_t N_BATCHED = (size_t)L_ * B_ * TOPK_ * LEN_ * D_;   // 2,457,600
  float* out_sb = out;
  float* out_tb = out_sb + N_BATCHED;
  float* out_ss = out_tb + N_BATCHED;                  // [64, 30]
  float* out_ts = out_ss + (size_t)B_ * POOL_;
  float* out_rs = out_ts + (size_t)B_ * POOL_;
  float* out_rt = out_rs + 1;

  // workspace layout (floats): partials | x_norm | s_key_norm(32) | t_key_norm(32) | idx
  float* ws   = (float*)d_ws;
  float* part = ws;                                       // SSPLIT_*B_*D_
  float* xn   = part + (size_t)SSPLIT_ * B_ * D_;         // B_*D_
  float* skn  = xn   + (size_t)B_ * D_;                   // POOLP_*D_
  float* tkn  = skn  + (size_t)POOLP_ * D_;               // POOLP_*D_
  int*   idxs = (int*)(tkn + (size_t)POOLP_ * D_);        // B_*TOPK_
  int*   idxt = idxs + B_ * TOPK_;                        // B_*TOPK_

  reduce_partial_kernel<<<dim3(B_, SSPLIT_), 192, 0, stream>>>(x_embed, part);
  finalize_xnorm_kernel<<<B_, 256, 0, stream>>>(part, xn);
  key_norm_kernel<<<dim3(2, POOLP_), 256, 0, stream>>>(s_pkey, t_pkey, skn, tkn);
  sim_wmma_kernel<<<dim3(4, 2, 2), 32, 0, stream>>>(xn, skn, tkn, out_ss, out_ts);
  topk_kernel<<<2, 64, 0, stream>>>(out_ss, out_ts, idxs, idxt, out_rs, out_rt);
  gather_kernel<<<dim3(B_ * TOPK_, L_, 2), 256, 0, stream>>>(
      s_prompt, t_prompt, idxs, idxt, out_sb, out_tb);
}